// LSTM_classifier2_26834955665541
// MI455X (gfx1250) — compile-verified
//
#include <hip/hip_runtime.h>

// Problem constants (match the reference)
#define BB   64     // batch
#define TT   256    // seq len
#define DD   512    // input size (layer 0)
#define HH   1024   // hidden size
#define PP   256    // projection size
#define G4H  4096   // 4*H

typedef float v2f __attribute__((ext_vector_type(2)));
typedef float v4f __attribute__((ext_vector_type(4)));
typedef float v8f __attribute__((ext_vector_type(8)));

// D = A(16x4) * B(4x16) + C, fp32 WMMA (CDNA5 V_WMMA_F32_16X16X4_F32)
__device__ __forceinline__ v8f wmma4(v2f a, v2f b, v8f c) {
    return __builtin_amdgcn_wmma_f32_16x16x4_f32(
        /*neg_a=*/false, a, /*neg_b=*/false, b,
        /*c_mod=*/(short)0, c, /*reuse_a=*/false, /*reuse_b=*/false);
}

// CDNA5 hardware transcendental tanh (V_TANH_F32) when available.
#if __has_builtin(__builtin_amdgcn_tanhf)
__device__ __forceinline__ float fast_tanh(float x) { return __builtin_amdgcn_tanhf(x); }
__device__ __forceinline__ float fast_sigmoid(float x) {
    return __builtin_amdgcn_tanhf(x * 0.5f) * 0.5f + 0.5f;
}
#else
__device__ __forceinline__ float fast_tanh(float x) { return tanhf(x); }
__device__ __forceinline__ float fast_sigmoid(float x) { return 1.0f / (1.0f + __expf(-x)); }
#endif

// ---------------------------------------------------------------------------
// Per-timestep gate + cell-update kernel, K split 2-ways across paired waves.
//   gates = x_t @ W_ih^T + hp_{t-1} @ W_hh^T + (b_ih + b_hh)
//   c = sig(f)*c + sig(i)*tanh(g);  hact = sig(o)*tanh(c)
// 512 waves: pair (2k,2k+1) shares one (16-row B-tile, 16-col H-tile) and
// splits the K reduction; partial tiles are reduced through LDS, the even
// wave applies bias + LSTM cell update. MODE 0: xin is [B,T,DIN]; MODE 1:
// xin is [T,B,DIN].
// ---------------------------------------------------------------------------
template <int DIN, int MODE>
__global__ __launch_bounds__(256) void gate_step(
    const float* __restrict__ xin,
    const float* __restrict__ hp,    // [B,P] previous projected hidden, or null at t==0
    const float* __restrict__ Wih,   // [4H, DIN]
    const float* __restrict__ Whh,   // [4H, P]
    const float* __restrict__ bias,  // [4H] = b_ih + b_hh
    float* __restrict__ cst,         // [B, H] cell state (in/out)
    float* __restrict__ hact,        // [B, H] o * tanh(c)
    int t)
{
    __shared__ float lds[4][4][256];  // [pair-in-block][gate][lane*8 + r] : 16 KB

    const int tid  = blockIdx.x * blockDim.x + threadIdx.x;
    const int gw   = tid >> 5;        // global wave 0..511
    const int lane = tid & 31;
    const int lo   = lane & 15;
    const int hi   = lane >> 4;
    const int ks   = gw & 1;          // K-split half
    const int mt   = (gw >> 1) & 3;   // B tile (B=64 -> 4 tiles)
    const int ht   = gw >> 3;         // H tile (H=1024 -> 64 tiles)
    const int pi   = (threadIdx.x >> 6); // pair index within block (0..3)
    const int m0   = mt * 16;
    const int n    = ht * 16 + lo;    // this lane's column within H

    v8f acc[4];
#pragma unroll
    for (int g = 0; g < 4; ++g) {
        const float bg = (ks == 0) ? bias[g * HH + n] : 0.0f;
#pragma unroll
        for (int r = 0; r < 8; ++r) acc[g][r] = bg;
    }

    // ---- x_t contribution (this wave's K half) ----------------------------
    const v2f* ap;
    if (MODE == 0)
        ap = (const v2f*)(xin + ((size_t)(m0 + lo) * TT + (size_t)t) * DIN);
    else
        ap = (const v2f*)(xin + ((size_t)t * BB + (size_t)(m0 + lo)) * DIN);

    const v2f* bp[4];
#pragma unroll
    for (int g = 0; g < 4; ++g)
        bp[g] = (const v2f*)(Wih + (size_t)(g * HH + n) * DIN);

    const int kx0 = ks * (DIN / 2), kx1 = kx0 + (DIN / 2);
#pragma unroll 4
    for (int k = kx0; k < kx1; k += 4) {
        const int o2 = (k >> 1) + hi;     // v2f index: floats k+2*hi, k+2*hi+1
        v2f a = ap[o2];
#pragma unroll
        for (int g = 0; g < 4; ++g)
            acc[g] = wmma4(a, bp[g][o2], acc[g]);
    }

    // ---- recurrent hp contribution (this wave's K half) -------------------
    if (hp != nullptr) {
        const v2f* hpp = (const v2f*)(hp + (size_t)(m0 + lo) * PP);
        const v2f* wp[4];
#pragma unroll
        for (int g = 0; g < 4; ++g)
            wp[g] = (const v2f*)(Whh + (size_t)(g * HH + n) * PP);
        const int kh0 = ks * (PP / 2), kh1 = kh0 + (PP / 2);
#pragma unroll 4
        for (int k = kh0; k < kh1; k += 4) {
            const int o2 = (k >> 1) + hi;
            v2f a = hpp[o2];
#pragma unroll
            for (int g = 0; g < 4; ++g)
                acc[g] = wmma4(a, wp[g][o2], acc[g]);
        }
    }

    // ---- pair reduction through LDS --------------------------------------
    if (ks == 1) {
#pragma unroll
        for (int g = 0; g < 4; ++g) {
            v4f l0, l1;
#pragma unroll
            for (int q = 0; q < 4; ++q) { l0[q] = acc[g][q]; l1[q] = acc[g][4 + q]; }
            *(v4f*)(&lds[pi][g][lane * 8])     = l0;
            *(v4f*)(&lds[pi][g][lane * 8 + 4]) = l1;
        }
    }
    __syncthreads();
    if (ks == 0) {
#pragma unroll
        for (int g = 0; g < 4; ++g) {
            v4f l0 = *(const v4f*)(&lds[pi][g][lane * 8]);
            v4f l1 = *(const v4f*)(&lds[pi][g][lane * 8 + 4]);
#pragma unroll
            for (int q = 0; q < 4; ++q) { acc[g][q] += l0[q]; acc[g][4 + q] += l1[q]; }
        }

        // ---- LSTM cell update: per-lane register math ---------------------
        // register r of a lane holds element (m0 + r + hi*8, n) of each tile
#pragma unroll
        for (int r = 0; r < 8; ++r) {
            const int m = m0 + r + hi * 8;
            const size_t idx = (size_t)m * HH + n;
            const float gi = fast_sigmoid(acc[0][r]);
            const float gf = fast_sigmoid(acc[1][r]);
            const float gg = fast_tanh(acc[2][r]);
            const float go = fast_sigmoid(acc[3][r]);
            const float cn = gf * cst[idx] + gi * gg;
            cst[idx]  = cn;
            hact[idx] = go * fast_tanh(cn);
        }
    }
}

// ---------------------------------------------------------------------------
// Projection: yout = hact[B,H] @ W_hr[P,H]^T -> [B,P].
// 128 waves: pair (2k,2k+1) shares one (B-tile, P-tile) and splits K=1024
// into 512-element halves; each wave runs two interleaved accumulator chains
// (chain length 64) to break the WMMA accumulation RAW dependency.
// ---------------------------------------------------------------------------
__global__ __launch_bounds__(256) void proj_step(
    const float* __restrict__ hact,  // [B, H]
    const float* __restrict__ Whr,   // [P, H]
    float* __restrict__ yout)        // [B, P]
{
    __shared__ float lds[4][256];    // [pair-in-block][lane*8 + r] : 4 KB

    const int tid  = blockIdx.x * blockDim.x + threadIdx.x;
    const int gw   = tid >> 5;        // global wave 0..127
    const int lane = tid & 31;
    const int lo   = lane & 15;
    const int hi   = lane >> 4;
    const int ks   = gw & 1;          // K-split half
    const int mt   = (gw >> 1) & 3;   // B tile
    const int pt   = gw >> 3;         // P tile (P=256 -> 16 tiles)
    const int pi   = (threadIdx.x >> 6);
    const int m0   = mt * 16;
    const int p0   = pt * 16;

    v8f acc0, acc1;
#pragma unroll
    for (int r = 0; r < 8; ++r) { acc0[r] = 0.0f; acc1[r] = 0.0f; }

    const v2f* ap = (const v2f*)(hact + (size_t)(m0 + lo) * HH);
    const v2f* bp = (const v2f*)(Whr + (size_t)(p0 + lo) * HH);

    const int k0 = ks * (HH / 2), k1 = k0 + (HH / 2);
#pragma unroll 4
    for (int k = k0; k < k1; k += 8) {
        const int oa = (k >> 1) + hi;
        const int ob = oa + 2;
        acc0 = wmma4(ap[oa], bp[oa], acc0);
        acc1 = wmma4(ap[ob], bp[ob], acc1);
    }
#pragma unroll
    for (int r = 0; r < 8; ++r) acc0[r] += acc1[r];

    if (ks == 1) {
        v4f l0, l1;
#pragma unroll
        for (int q = 0; q < 4; ++q) { l0[q] = acc0[q]; l1[q] = acc0[4 + q]; }
        *(v4f*)(&lds[pi][lane * 8])     = l0;
        *(v4f*)(&lds[pi][lane * 8 + 4]) = l1;
    }
    __syncthreads();
    if (ks == 0) {
        v4f l0 = *(const v4f*)(&lds[pi][lane * 8]);
        v4f l1 = *(const v4f*)(&lds[pi][lane * 8 + 4]);
#pragma unroll
        for (int q = 0; q < 4; ++q) { acc0[q] += l0[q]; acc0[4 + q] += l1[q]; }
#pragma unroll
        for (int r = 0; r < 8; ++r) {
            const int m = m0 + r + hi * 8;
            yout[(size_t)m * PP + p0 + lo] = acc0[r];
        }
    }
}

__global__ void add_bias(const float* __restrict__ a, const float* __restrict__ b,
                         float* __restrict__ o, int n)
{
    int i = blockIdx.x * blockDim.x + threadIdx.x;
    if (i < n) o[i] = a[i] + b[i];
}

// ---------------------------------------------------------------------------
extern "C" void kernel_launch(void* const* d_in, const int* in_sizes, int n_in,
                              void* d_out, int out_size, void* d_ws, size_t ws_size,
                              hipStream_t stream)
{
    (void)in_sizes; (void)n_in; (void)out_size; (void)ws_size;

    const float* x    = (const float*)d_in[0];
    const float* Wih0 = (const float*)d_in[1];
    const float* Whh0 = (const float*)d_in[2];
    const float* bih0 = (const float*)d_in[3];
    const float* bhh0 = (const float*)d_in[4];
    const float* Whr0 = (const float*)d_in[5];
    const float* Wihs = (const float*)d_in[6];
    const float* Whhs = (const float*)d_in[7];
    const float* bihs = (const float*)d_in[8];
    const float* bhhs = (const float*)d_in[9];
    const float* Whrs = (const float*)d_in[10];

    // workspace layout (~33 MB, L2-resident)
    float* ws   = (float*)d_ws;
    float* bias = ws;                              // 4 * 4096
    float* cbuf = bias + 4 * G4H;                  // 64 * 1024
    float* hbuf = cbuf + (size_t)BB * HH;          // 64 * 1024
    float* Ya   = hbuf + (size_t)BB * HH;          // T*B*P
    float* Yb   = Ya + (size_t)TT * BB * PP;       // T*B*P

    // fused biases per layer
    for (int l = 0; l < 4; ++l) {
        const float* bi = (l == 0) ? bih0 : bihs + (size_t)(l - 1) * G4H;
        const float* bh = (l == 0) ? bhh0 : bhhs + (size_t)(l - 1) * G4H;
        add_bias<<<(G4H + 255) / 256, 256, 0, stream>>>(bi, bh, bias + (size_t)l * G4H, G4H);
    }

    const float* yin = nullptr;   // previous layer's output sequence [T,B,P]
    float* yout = Ya;
    for (int l = 0; l < 4; ++l) {
        const float* Wih = (l == 0) ? Wih0 : Wihs + (size_t)(l - 1) * G4H * PP;
        const float* Whh = (l == 0) ? Whh0 : Whhs + (size_t)(l - 1) * G4H * PP;
        const float* Whr = (l == 0) ? Whr0 : Whrs + (size_t)(l - 1) * PP * HH;
        const float* bl  = bias + (size_t)l * G4H;

        hipMemsetAsync(cbuf, 0, (size_t)BB * HH * sizeof(float), stream);

        for (int t = 0; t < TT; ++t) {
            const float* hp = (t == 0) ? nullptr : yout + (size_t)(t - 1) * BB * PP;
            if (l == 0)
                gate_step<DD, 0><<<64, 256, 0, stream>>>(x,   hp, Wih, Whh, bl, cbuf, hbuf, t);
            else
                gate_step<PP, 1><<<64, 256, 0, stream>>>(yin, hp, Wih, Whh, bl, cbuf, hbuf, t);
            proj_step<<<16, 256, 0, stream>>>(hbuf, Whr, yout + (size_t)t * BB * PP);
        }
        yin  = yout;
        yout = (yout == Ya) ? Yb : Ya;
    }

    // final output = last layer's hp at t = T-1
    hipMemcpyAsync(d_out, (const void*)(yin + (size_t)(TT - 1) * BB * PP),
                   (size_t)BB * PP * sizeof(float), hipMemcpyDeviceToDevice, stream);
}